// BaselineGCN_21586505630436
// MI455X (gfx1250) — compile-verified
//
#include <hip/hip_runtime.h>

typedef __attribute__((ext_vector_type(2))) float v2f;
typedef __attribute__((ext_vector_type(8))) float v8f;

#define HID 64
#define NGRAPH 256

// ---------------- utility fills ----------------
__global__ void k_fill(float* __restrict__ p, float v, int n) {
  int i = blockIdx.x * blockDim.x + threadIdx.x;
  if (i < n) p[i] = v;
}

// ---------------- degree / dinv (computed once, reused for all 3 layers) ---
__global__ void k_deg(const int* __restrict__ col, float* __restrict__ deg, int E) {
  int e = blockIdx.x * blockDim.x + threadIdx.x;
  if (e < E) atomicAdd(&deg[col[e]], 1.0f);
}

__global__ void k_dinv(float* __restrict__ deg, int n) {
  int i = blockIdx.x * blockDim.x + threadIdx.x;
  if (i < n) deg[i] = rsqrtf(deg[i]);   // deg >= 1 (self loop folded into init)
}

// ---------------- GEMM: Out[N,64] = A[N,K] @ Bw[K,64] via V_WMMA_F32_16X16X4_F32
// One wave computes a 16x64 strip (4 N-tiles), K-step 4, f32 end-to-end.
__global__ __launch_bounds__(128)
void k_gemm_wmma(const float* __restrict__ A, const float* __restrict__ Bw,
                 float* __restrict__ Out, int K, int mtiles) {
  const int wave = threadIdx.x >> 5;
  const int lane = threadIdx.x & 31;
  const int tile = blockIdx.x * 4 + wave;
  if (tile >= mtiles) return;            // wave-uniform: EXEC stays all-1s

  const int half = lane >> 4;            // 0: K pair {0,1}, 1: K pair {2,3}
  const int l15  = lane & 15;
  const int m0   = tile * 16;

  v8f c0 = {}, c1 = {}, c2 = {}, c3 = {};
  const float* arow = A + (size_t)(m0 + l15) * K;

  for (int k0 = 0; k0 < K; k0 += 4) {
    // A-fragment: 16x4 f32 (ISA 7.12.2 layout)
    v2f a = *(const v2f*)(arow + k0 + 2 * half);
    // B-fragments: 4x16 f32, column = nt*16 + l15, rows k0+2*half .. +1
    const float* bp = Bw + (size_t)(k0 + 2 * half) * HID + l15;
    v2f b0, b1, b2, b3;
    b0.x = bp[0];  b0.y = bp[HID + 0];
    b1.x = bp[16]; b1.y = bp[HID + 16];
    b2.x = bp[32]; b2.y = bp[HID + 32];
    b3.x = bp[48]; b3.y = bp[HID + 48];
    c0 = __builtin_amdgcn_wmma_f32_16x16x4_f32(false, a, false, b0, (short)0, c0, false, false);
    c1 = __builtin_amdgcn_wmma_f32_16x16x4_f32(false, a, false, b1, (short)0, c1, false, false);
    c2 = __builtin_amdgcn_wmma_f32_16x16x4_f32(false, a, false, b2, (short)0, c2, false, false);
    c3 = __builtin_amdgcn_wmma_f32_16x16x4_f32(false, a, false, b3, (short)0, c3, false, false);
  }

  // C/D layout: VGPR r -> row m0 + 8*half + r, column l15 (+16*nt)
  float* orow = Out + (size_t)(m0 + 8 * half) * HID + l15;
  #pragma unroll
  for (int r = 0; r < 8; ++r) {
    orow[(size_t)r * HID + 0]  = c0[r];
    orow[(size_t)r * HID + 16] = c1[r];
    orow[(size_t)r * HID + 32] = c2[r];
    orow[(size_t)r * HID + 48] = c3[r];
  }
}

// ---------------- self-loop init: out[v,f] = dinv[v]^2 * h[v,f] -------------
__global__ void k_self_init(const float* __restrict__ h, const float* __restrict__ dinv,
                            float* __restrict__ out, int n) {
  int i = blockIdx.x * blockDim.x + threadIdx.x;
  if (i < n) {
    float d = dinv[i >> 6];
    out[i] = d * d * h[i];
  }
}

// ---------------- edge scatter: out[c,f] += dinv[r]*dinv[c]*h[r,f] ----------
// 64 lanes per edge -> coalesced gather of h[r,:] and coalesced f32 atomics.
__global__ void k_edge_agg(const int* __restrict__ ei, const float* __restrict__ dinv,
                           const float* __restrict__ h, float* __restrict__ out,
                           int E, int total) {
  int idx = blockIdx.x * blockDim.x + threadIdx.x;
  if (idx >= total) return;
  int e = idx >> 6, f = idx & 63;
  int r = ei[e];
  int c = ei[E + e];
  float w = dinv[r] * dinv[c];
  atomicAdd(&out[(size_t)c * HID + f], w * h[(size_t)r * HID + f]);
}

// ---------------- bias + relu in place --------------------------------------
__global__ void k_bias_relu(float* __restrict__ out, const float* __restrict__ bias, int n) {
  int i = blockIdx.x * blockDim.x + threadIdx.x;
  if (i < n) {
    float v = out[i] + bias[i & 63];
    out[i] = v > 0.0f ? v : 0.0f;
  }
}

// ---------------- global mean pool accumulation -----------------------------
__global__ void k_pool(const float* __restrict__ h, const int* __restrict__ batch,
                       float* __restrict__ gsum, float* __restrict__ gcnt, int n) {
  int i = blockIdx.x * blockDim.x + threadIdx.x;
  if (i < n) {
    int v = i >> 6, f = i & 63;
    int b = batch[v];
    atomicAdd(&gsum[(size_t)b * HID + f], h[i]);
    if (f == 0) atomicAdd(&gcnt[b], 1.0f);
  }
}

// ---------------- final MLP: out = (mean @ Wl1 + bl1) @ Wl2 + bl2 -----------
__global__ void k_mlp(const float* __restrict__ gsum, const float* __restrict__ gcnt,
                      const float* __restrict__ Wl1, const float* __restrict__ bl1,
                      const float* __restrict__ Wl2, const float* __restrict__ bl2,
                      float* __restrict__ out) {
  int g = blockIdx.x * blockDim.x + threadIdx.x;
  if (g >= NGRAPH) return;
  float inv = 1.0f / fmaxf(gcnt[g], 1.0f);
  float hid[16];
  #pragma unroll
  for (int j = 0; j < 16; ++j) hid[j] = bl1[j];
  for (int k = 0; k < HID; ++k) {
    float x = gsum[(size_t)g * HID + k] * inv;
    #pragma unroll
    for (int j = 0; j < 16; ++j) hid[j] = fmaf(x, Wl1[k * 16 + j], hid[j]);
  }
  float o = bl2[0];
  #pragma unroll
  for (int j = 0; j < 16; ++j) o = fmaf(hid[j], Wl2[j], o);
  out[g] = o;
}

extern "C" void kernel_launch(void* const* d_in, const int* in_sizes, int n_in,
                              void* d_out, int out_size, void* d_ws, size_t ws_size,
                              hipStream_t stream) {
  const float* x    = (const float*)d_in[0];
  const int*   ei   = (const int*)  d_in[1];   // [2, E] flattened
  const int*   batch= (const int*)  d_in[2];
  const float* W1   = (const float*)d_in[3];
  const float* b1   = (const float*)d_in[4];
  const float* W2   = (const float*)d_in[5];
  const float* b2   = (const float*)d_in[6];
  const float* W3   = (const float*)d_in[7];
  const float* b3   = (const float*)d_in[8];
  const float* Wl1  = (const float*)d_in[9];
  const float* bl1  = (const float*)d_in[10];
  const float* Wl2  = (const float*)d_in[11];
  const float* bl2  = (const float*)d_in[12];
  float* out = (float*)d_out;

  const int IN_DIM = 128;
  const int N = in_sizes[0] / IN_DIM;          // 50000
  const int E = in_sizes[1] / 2;               // 1600000
  const int NF = N * HID;                      // node-feature elements
  const int ET = E * HID;                      // edge x feature threads

  // workspace layout (floats)
  float* ws   = (float*)d_ws;
  float* dinv = ws;                            // N (deg -> dinv in place)
  float* hA   = ws + ((N + 63) & ~63);         // N*64 (GEMM output)
  float* hB   = hA + (size_t)NF;               // N*64 (aggregated / layer input)
  float* gsum = hB + (size_t)NF;               // 256*64
  float* gcnt = gsum + (size_t)NGRAPH * HID;   // 256

  const int T = 256;
  dim3 blk(T);
  const int mtiles = (N + 15) / 16;            // 3125 (exact)
  dim3 gemmGrid((mtiles + 3) / 4), gemmBlk(128);

  // degree (self-loop => init 1.0), then dinv = rsqrt(deg)   [once]
  k_fill<<<(N + T - 1) / T, blk, 0, stream>>>(dinv, 1.0f, N);
  k_deg <<<(E + T - 1) / T, blk, 0, stream>>>(ei + E, dinv, E);
  k_dinv<<<(N + T - 1) / T, blk, 0, stream>>>(dinv, N);

  // ---- layer 1: K = 128, input x ----
  k_gemm_wmma<<<gemmGrid, gemmBlk, 0, stream>>>(x, W1, hA, IN_DIM, mtiles);
  k_self_init<<<(NF + T - 1) / T, blk, 0, stream>>>(hA, dinv, hB, NF);
  k_edge_agg <<<(ET + T - 1) / T, blk, 0, stream>>>(ei, dinv, hA, hB, E, ET);
  k_bias_relu<<<(NF + T - 1) / T, blk, 0, stream>>>(hB, b1, NF);

  // ---- layer 2: K = 64, input hB ----
  k_gemm_wmma<<<gemmGrid, gemmBlk, 0, stream>>>(hB, W2, hA, HID, mtiles);
  k_self_init<<<(NF + T - 1) / T, blk, 0, stream>>>(hA, dinv, hB, NF);
  k_edge_agg <<<(ET + T - 1) / T, blk, 0, stream>>>(ei, dinv, hA, hB, E, ET);
  k_bias_relu<<<(NF + T - 1) / T, blk, 0, stream>>>(hB, b2, NF);

  // ---- layer 3: K = 64, input hB ----
  k_gemm_wmma<<<gemmGrid, gemmBlk, 0, stream>>>(hB, W3, hA, HID, mtiles);
  k_self_init<<<(NF + T - 1) / T, blk, 0, stream>>>(hA, dinv, hB, NF);
  k_edge_agg <<<(ET + T - 1) / T, blk, 0, stream>>>(ei, dinv, hA, hB, E, ET);
  k_bias_relu<<<(NF + T - 1) / T, blk, 0, stream>>>(hB, b3, NF);

  // ---- mean pool + MLP head ----
  k_fill<<<(NGRAPH * HID + NGRAPH + T - 1) / T, blk, 0, stream>>>(gsum, 0.0f, NGRAPH * HID + NGRAPH);
  k_pool<<<(NF + T - 1) / T, blk, 0, stream>>>(hB, batch, gsum, gcnt, NF);
  k_mlp <<<1, NGRAPH, 0, stream>>>(gsum, gcnt, Wl1, bl1, Wl2, bl2, out);
}